// GCN_40544491274720
// MI455X (gfx1250) — compile-verified
//
#include <hip/hip_runtime.h>

typedef float v2f __attribute__((ext_vector_type(2)));
typedef float v8f __attribute__((ext_vector_type(8)));

#define NNODES 50000
#define HDIM   128

// Force the native global_atomic_add_f32 (no CAS loop). unsafeAtomicAdd is the
// HIP wrapper for __builtin_amdgcn_global_atomic_fadd_f32 on AMD targets.
__device__ __forceinline__ void atomic_fadd(float* p, float v) {
#if defined(__HIP_DEVICE_COMPILE__) && defined(__AMDGCN__)
  unsafeAtomicAdd(p, v);
#else
  __hip_atomic_fetch_add(p, v, __ATOMIC_RELAXED, __HIP_MEMORY_SCOPE_AGENT);
#endif
}

// ---------------- degree / normalization ----------------

__global__ void deg_init_kernel(float* __restrict__ deg, int n) {
  int i = blockIdx.x * blockDim.x + threadIdx.x;
  if (i < n) deg[i] = 1.0f;  // self-loop contributes 1 to every node's degree
}

__global__ void deg_count_kernel(const long long* __restrict__ dst,
                                 float* __restrict__ deg, int e) {
  int i = blockIdx.x * blockDim.x + threadIdx.x;
  if (i < e) {
    atomic_fadd(&deg[dst[i]], 1.0f);
  }
}

__global__ void dinv_kernel(float* __restrict__ d, int n) {
  int i = blockIdx.x * blockDim.x + threadIdx.x;
  if (i < n) {
    float v = d[i];
    d[i] = (v > 0.0f) ? rsqrtf(v) : 0.0f;
  }
}

// ---------------- WMMA GEMM: C[M x NC] = A[M x 128] @ W[128 x NC] (+bias) ----
// block = 128 threads = 4 waves; each wave computes a 16-row M tile.

template <int NC>
__launch_bounds__(128)
__global__ void gemm_xw_kernel(const float* __restrict__ A,
                               const float* __restrict__ W,
                               const float* __restrict__ bias,  // may be null
                               float* __restrict__ C, int M) {
  __shared__ float Wl[128 * NC];

  const int tid = threadIdx.x;
  // cooperative load of W into LDS (row-major [k][n]) — fully coalesced
  for (int i = tid; i < 128 * NC; i += 128) Wl[i] = W[i];
  __syncthreads();

  const int wave = tid >> 5;
  const int lane = tid & 31;
  const int row16 = (blockIdx.x * 4 + wave) * 16;
  if (row16 >= M) return;

  const int arow  = row16 + (lane & 15);
  const int khalf = (lane >> 4) * 2;  // 0 for lanes 0-15, 2 for lanes 16-31

  // Preload this wave's 16x128 A tile as WMMA fragments:
  // lane holds A[arow][4s+khalf], A[arow][4s+khalf+1] for s = 0..31
  const float* arow_ptr = A + (size_t)arow * 128;
  float2 afrag[32];
#pragma unroll
  for (int s = 0; s < 32; ++s)
    afrag[s] = *(const float2*)(arow_ptr + 4 * s + khalf);

#pragma unroll
  for (int nt = 0; nt < NC / 16; ++nt) {
    const int col = nt * 16 + (lane & 15);
    v8f acc = {};
#pragma unroll
    for (int s = 0; s < 32; ++s) {
      v2f a;
      a.x = afrag[s].x;
      a.y = afrag[s].y;
      v2f b;  // B 4x16 fragment: lane holds W[4s+khalf][col], W[4s+khalf+1][col]
      b.x = Wl[(4 * s + khalf) * NC + col];
      b.y = Wl[(4 * s + khalf + 1) * NC + col];
      acc = __builtin_amdgcn_wmma_f32_16x16x4_f32(
          /*neg_a=*/false, a, /*neg_b=*/false, b,
          /*c_mod=*/(short)0, acc, /*reuse_a=*/false, /*reuse_b=*/false);
    }
    const float bv = bias ? bias[col] : 0.0f;
    // C/D layout: VGPR r -> M = r (lanes 0-15) / M = r+8 (lanes 16-31)
    const int crowbase = row16 + (lane >> 4) * 8;
#pragma unroll
    for (int r = 0; r < 8; ++r)
      C[(size_t)(crowbase + r) * NC + col] = acc[r] + bv;
  }
}

// ---------------- aggregation: self-loop init, edge scatter, bias+relu ------

__global__ void self_init_kernel(const float* __restrict__ t,
                                 const float* __restrict__ dinv,
                                 float* __restrict__ out, int n) {
  int i = blockIdx.x * blockDim.x + threadIdx.x;  // over n * (HDIM/4)
  if (i >= n * (HDIM / 4)) return;
  int node = i >> 5;
  int f = (i & 31) * 4;
  float s = dinv[node];
  s = s * s;
  float4 v = *(const float4*)(t + (size_t)node * HDIM + f);
  float4 o;
  o.x = v.x * s; o.y = v.y * s; o.z = v.z * s; o.w = v.w * s;
  *(float4*)(out + (size_t)node * HDIM + f) = o;
}

__global__ void scatter_kernel(const float* __restrict__ t,
                               const long long* __restrict__ src,
                               const long long* __restrict__ dst,
                               const float* __restrict__ dinv,
                               float* __restrict__ out, int e) {
  long long i = (long long)blockIdx.x * blockDim.x + threadIdx.x;
  if (i >= (long long)e * 32) return;
  int edge = (int)(i >> 5);
  int f = (int)(i & 31) * 4;
  long long s = src[edge];
  long long d = dst[edge];
  float nrm = dinv[s] * dinv[d];
  float4 v = *(const float4*)(t + s * HDIM + f);
  float* o = out + d * HDIM + f;
  atomic_fadd(o + 0, v.x * nrm);
  atomic_fadd(o + 1, v.y * nrm);
  atomic_fadd(o + 2, v.z * nrm);
  atomic_fadd(o + 3, v.w * nrm);
}

__global__ void bias_relu_kernel(float* __restrict__ out,
                                 const float* __restrict__ b, int n) {
  int i = blockIdx.x * blockDim.x + threadIdx.x;  // over n * (HDIM/4)
  if (i >= n * (HDIM / 4)) return;
  int f = (i & 31) * 4;
  float4 v = *(const float4*)(out + (size_t)i * 4);
  float4 bb = *(const float4*)(b + f);
  v.x = fmaxf(v.x + bb.x, 0.0f);
  v.y = fmaxf(v.y + bb.y, 0.0f);
  v.z = fmaxf(v.z + bb.z, 0.0f);
  v.w = fmaxf(v.w + bb.w, 0.0f);
  *(float4*)(out + (size_t)i * 4) = v;
}

// ---------------- driver ----------------

extern "C" void kernel_launch(void* const* d_in, const int* in_sizes, int n_in,
                              void* d_out, int out_size, void* d_ws, size_t ws_size,
                              hipStream_t stream) {
  const float* x  = (const float*)d_in[0];
  const long long* edge_index = (const long long*)d_in[1];
  const float* W1 = (const float*)d_in[2];
  const float* b1 = (const float*)d_in[3];
  const float* W2 = (const float*)d_in[4];
  const float* b2 = (const float*)d_in[5];
  const float* Wc = (const float*)d_in[6];
  const float* bc = (const float*)d_in[7];
  float* out = (float*)d_out;

  const int n = NNODES;
  const int e = in_sizes[1] / 2;  // edge_index is (2, E) int64
  const long long* src = edge_index;
  const long long* dst = edge_index + e;

  // workspace: [dinv: n floats | t: n*H | h: n*H]
  char* ws = (char*)d_ws;
  float* dinv = (float*)ws;
  float* t = (float*)(ws + (((size_t)n * sizeof(float) + 255) & ~(size_t)255));
  float* h = t + (size_t)n * HDIM;

  const int THREADS = 256;
  const int nb_n   = (n + THREADS - 1) / THREADS;
  const int nb_e   = (e + THREADS - 1) / THREADS;
  const int nb_nf  = (n * (HDIM / 4) + THREADS - 1) / THREADS;
  const long long escatter = (long long)e * 32;
  const int nb_es  = (int)((escatter + THREADS - 1) / THREADS);
  const int nb_gemm = ((n / 16) + 3) / 4;  // 4 waves (16-row tiles) per block

  // normalization: deg (with self-loops) -> d^{-1/2}
  deg_init_kernel<<<nb_n, THREADS, 0, stream>>>(dinv, n);
  deg_count_kernel<<<nb_e, THREADS, 0, stream>>>(dst, dinv, e);
  dinv_kernel<<<nb_n, THREADS, 0, stream>>>(dinv, n);

  // layer 1: t = x @ W1 ; h = scatter(norm * t) + b1 ; relu
  gemm_xw_kernel<128><<<nb_gemm, 128, 0, stream>>>(x, W1, nullptr, t, n);
  self_init_kernel<<<nb_nf, THREADS, 0, stream>>>(t, dinv, h, n);
  scatter_kernel<<<nb_es, THREADS, 0, stream>>>(t, src, dst, dinv, h, e);
  bias_relu_kernel<<<nb_nf, THREADS, 0, stream>>>(h, b1, n);

  // layer 2: t = h @ W2 ; h = scatter(norm * t) + b2 ; relu
  gemm_xw_kernel<128><<<nb_gemm, 128, 0, stream>>>(h, W2, nullptr, t, n);
  self_init_kernel<<<nb_nf, THREADS, 0, stream>>>(t, dinv, h, n);
  scatter_kernel<<<nb_es, THREADS, 0, stream>>>(t, src, dst, dinv, h, e);
  bias_relu_kernel<<<nb_nf, THREADS, 0, stream>>>(h, b2, n);

  // classifier: out = h @ Wc + bc
  gemm_xw_kernel<16><<<nb_gemm, 128, 0, stream>>>(h, Wc, bc, out, n);
}